// HeatEquation2M_45578192945900
// MI455X (gfx1250) — compile-verified
//
#include <hip/hip_runtime.h>

// ---------------------------------------------------------------------------
// 2-material heat equation, 512x512, 499 emitted states.
//
// Roofline (MI455X): 523 MB compulsory output stores -> ~22us at 23.3 TB/s.
// State (1 MB) is L2-resident (192 MB L2). The cost driver is the 499-step
// serial chain, so: ONE persistent launch, temporal blocking (8 steps/chunk
// -> 62 device-wide barriers), tiles staged into LDS with CDNA5 async
// global->LDS b128 loads (ASYNCcnt path), column-invariant compute with
// rolling registers. WMMA deliberately unused: a radius-1 stencil at
// ~1 GFLOP total has no matmul intensity; async-tensor data movement is the
// CDNA5 feature that matters here.
// ---------------------------------------------------------------------------

#define NN      512
#define NSTEPS  499                 // emitted states (steps 1..499)
#define MBND    256                 // material boundary column count
#define TILE    32                  // core tile per workgroup
#define CH      8                   // temporal chunk (steps per grid sync)
#define HT      (TILE + 2 * CH)     // 48: staged tile incl. halo
#define HTP     56                  // padded LDS row stride (floats): 16B
                                    // multiple AND bank-conflict-free
#define NWGX    (NN / TILE)         // 16
#define NWG     (NWGX * NWGX)       // 256 workgroups
#define BLK     288                 // 9 wave32 waves: 48 cols x 6 row-strips

// CDNA5 async global->LDS, 16 bytes. VDST is a VGPR holding the LDS byte
// offset; generic pointers into LDS carry the offset in their low 32 bits.
__device__ __forceinline__ void async_copy_b128(const float* g, float* l) {
    unsigned lds_off = (unsigned)(unsigned long long)l;
    asm volatile("global_load_async_to_lds_b128 %0, %1, off"
                 :
                 : "v"(lds_off), "v"(g)
                 : "memory");
}

__device__ __forceinline__ void wait_async0() {
    asm volatile("s_wait_asynccnt 0" ::: "memory");
}

__global__ __launch_bounds__(BLK)
void HeatEquation2M_kernel(const float* __restrict__ u0,
                           const float* kp1, const float* kp2,
                           const float* ap1, const float* ap2,
                           float* __restrict__ out,
                           unsigned* __restrict__ barrier_ctr)
{
    __shared__ __align__(16) float buf0[HT][HTP];
    __shared__ __align__(16) float buf1[HT][HTP];

    const int tid  = threadIdx.x;
    const int bx   = blockIdx.x;
    const int by   = blockIdx.y;
    const int row0 = by * TILE - CH;   // global row of tile-local row 0
    const int col0 = bx * TILE - CH;   // global col of tile-local col 0

    // Uniform scalar parameters (single-element input arrays).
    const float k1 = kp1[0];
    const float k2 = kp2[0];
    const float rs = 1.0f / (k1 + k2);
    const float w_right = k1 * rs;     // multiplies old T[:, 256]
    const float w_left  = k2 * rs;     // multiplies old T[:, 254]
    const float dx      = 1.0f / (float)(NN - 1);
    const float inv_dx2 = 1.0f / (dx * dx);
    const float dt      = 5e-7f;
    const float coefA   = dt * ap1[0] * inv_dx2;   // columns < MBND
    const float coefB   = dt * ap2[0] * inv_dx2;   // columns >= MBND

    // ---- Per-thread loop-invariant column state (hoisted out of the 499-
    // step hot path): fixed column jj, 8-row strip [rstart, rend).
    const int  jj     = tid % 48;
    const int  rblk   = tid / 48;              // 0..5
    const int  rstart = 1 + rblk * 8;          // 1,9,17,25,33,41
    const int  rend   = (rstart + 8 < HT - 1) ? (rstart + 8) : (HT - 1);
    const int  gj     = col0 + jj;
    const bool col_active = (jj >= 1) && (jj <= HT - 2);
    const bool col_zero   = ((unsigned)(gj - 1) >= (unsigned)(NN - 2));
    const bool is_iface   = (gj == MBND - 1);
    const float coef      = (gj < MBND) ? coefA : coefB;

    // Hygiene: zero buf1 once so never-written tile edges are finite.
    for (int idx = tid; idx < HT * HTP; idx += BLK)
        (&buf1[0][0])[idx] = 0.0f;
    __syncthreads();

    int chunk = 0;
    for (int t = 0; t < NSTEPS; t += CH, ++chunk) {
        const int cs = (NSTEPS - t < CH) ? (NSTEPS - t) : CH;
        // Global state after t steps: u0 for t==0, else output slice t-1.
        const float* src = (t == 0) ? u0 : (out + (size_t)(t - 1) * NN * NN);

        // ---- Stage 48x48 tile as 576 16-byte spans (exactly 2 per thread).
        // Tile columns are 4-aligned and domain edges are multiples of 4, so
        // every span is fully in-domain or fully ghost.
        for (int p = 0; p < 2; ++p) {
            int s  = tid + p * BLK;            // 0..575
            int ii = s / 12;                   // tile row
            int c4 = (s - ii * 12) * 4;        // tile col (multiple of 4)
            int gi = row0 + ii;
            int gc = col0 + c4;
            if ((unsigned)gi < (unsigned)NN && (unsigned)gc < (unsigned)NN) {
                async_copy_b128(src + (size_t)gi * NN + gc, &buf0[ii][c4]);
            } else {
                float4 z; z.x = z.y = z.z = z.w = 0.0f;
                *(float4*)&buf0[ii][c4] = z;   // ghost outside the domain
            }
        }
        wait_async0();
        __syncthreads();

        float (*cur)[HTP] = buf0;
        float (*nxt)[HTP] = buf1;

        // ---- cs local steps in LDS; valid window shrinks 1 cell/step and
        // halo CH=8 exactly protects the 32x32 core [8,40).
        for (int k = 1; k <= cs; ++k) {
            if (col_active) {
                float a = cur[rstart - 1][jj];      // row above strip
                float b = cur[rstart][jj];          // current row
                for (int ii = rstart; ii < rend; ++ii) {
                    float dn = cur[ii + 1][jj];
                    float lf = cur[ii][jj - 1];
                    float rt = cur[ii][jj + 1];
                    int   gi = row0 + ii;
                    float v;
                    if (col_zero || (unsigned)(gi - 1) >= (unsigned)(NN - 2)) {
                        v = 0.0f;                   // Dirichlet edge / ghost
                    } else if (is_iface) {
                        v = w_right * rt + w_left * lf;  // from OLD state
                    } else {
                        v = b + coef * (a + dn + lf + rt - 4.0f * b);
                    }
                    nxt[ii][jj] = v;
                    a = b; b = dn;                  // roll the 3-row window
                }
            }
            __syncthreads();

            // ---- Emit this step's 32x32 core to output slice (t+k-1).
            if (tid < 256) {
                float* o = out + (size_t)(t + k - 1) * NN * NN;
                int r  = tid >> 3;                  // 0..31 core row
                int c4 = (tid & 7) << 2;            // 0,4,..,28 core col
                const float4 v4 = *(const float4*)&nxt[r + CH][c4 + CH];
                *(float4*)(o + (size_t)(by * TILE + r) * NN + (bx * TILE + c4)) = v4;
            }

            float (*tmp)[HTP] = cur; cur = nxt; nxt = tmp;
        }

        // ---- Device-wide barrier between chunks (skip after the last one).
        if (t + CH < NSTEPS) {
            __builtin_amdgcn_fence(__ATOMIC_RELEASE, "agent");
            __syncthreads();
            if (tid == 0) {
                __hip_atomic_fetch_add(barrier_ctr, 1u, __ATOMIC_RELAXED,
                                       __HIP_MEMORY_SCOPE_AGENT);
                unsigned target = (unsigned)(chunk + 1) * (unsigned)NWG;
                long spins = 0;
                while (__hip_atomic_load(barrier_ctr, __ATOMIC_RELAXED,
                                         __HIP_MEMORY_SCOPE_AGENT) < target) {
                    __builtin_amdgcn_s_sleep(2);
                    if (++spins > (1L << 26)) break;   // safety valve
                }
            }
            __syncthreads();
            __builtin_amdgcn_fence(__ATOMIC_ACQUIRE, "agent");
        }
    }
}

extern "C" void kernel_launch(void* const* d_in, const int* in_sizes, int n_in,
                              void* d_out, int out_size, void* d_ws, size_t ws_size,
                              hipStream_t stream) {
    (void)in_sizes; (void)n_in; (void)out_size;
    const float* u0 = (const float*)d_in[0];
    const float* k1 = (const float*)d_in[1];
    const float* k2 = (const float*)d_in[2];
    const float* a1 = (const float*)d_in[3];
    const float* a2 = (const float*)d_in[4];
    float*       out = (float*)d_out;
    unsigned*    ctr = (unsigned*)d_ws;

    // Barrier counter must start at 0 every call (ws is poisoned, not zeroed).
    size_t clr = ws_size < 64 ? ws_size : 64;
    hipMemsetAsync(d_ws, 0, clr, stream);

    dim3 grid(NWGX, NWGX);   // 16 x 16 = 256 persistent workgroups
    HeatEquation2M_kernel<<<grid, BLK, 0, stream>>>(u0, k1, k2, a1, a2, out, ctr);
}